// Net_89584427860039
// MI455X (gfx1250) — compile-verified
//
#include <hip/hip_runtime.h>
#include <math.h>

typedef float v2f __attribute__((ext_vector_type(2)));
typedef float v8f __attribute__((ext_vector_type(8)));

#define HID 16
#define CLS 8

// ---------------- kernel: zero u32 counts ----------------
__global__ void k_zero_u32(unsigned* __restrict__ p, int n) {
  int i = blockIdx.x * blockDim.x + threadIdx.x;
  if (i < n) p[i] = 0u;
}

// ---------------- kernel: count edges per (relation, dst) ----------------
__global__ void k_count(const int* __restrict__ dst, const int* __restrict__ et,
                        unsigned* __restrict__ cnt, int E, int N) {
  int e = blockIdx.x * blockDim.x + threadIdx.x;
  if (e >= E) return;
  int d = dst[e];
  int r = et[e];
  atomicAdd(&cnt[(size_t)r * N + d], 1u);
}

// ---------------- kernel: cnt -> 1/max(cnt,1), in place ----------------
__global__ void k_invcnt(float* __restrict__ p, int n) {
  int i = blockIdx.x * blockDim.x + threadIdx.x;
  if (i >= n) return;
  unsigned c = reinterpret_cast<unsigned*>(p)[i];
  p[i] = 1.0f / (float)(c ? c : 1u);
}

// ---------------- kernel: h = root1 + bias1 ----------------
__global__ void k_init_h(float* __restrict__ h, const float* __restrict__ root1,
                         const float* __restrict__ bias1, int NH) {
  int i = blockIdx.x * blockDim.x + threadIdx.x;
  if (i >= NH) return;
  h[i] = root1[i] + bias1[i & (HID - 1)];
}

// ---------------- kernel: layer-1 edge pass ----------------
// h[d] += weight1[r, src] * inv[r, d]   (mean folded into per-edge scale)
__global__ void k_edge1(const int* __restrict__ src, const int* __restrict__ dst,
                        const int* __restrict__ et, const float* __restrict__ w1,
                        const float* __restrict__ inv, float* __restrict__ h,
                        int E, int N) {
  int e = blockIdx.x * blockDim.x + threadIdx.x;
  if (e >= E) return;
  int s = src[e], d = dst[e], r = et[e];
  float iv = inv[(size_t)r * N + d];
  const float4* w = reinterpret_cast<const float4*>(&w1[((size_t)r * N + s) * HID]);
  float* hd = &h[(size_t)d * HID];
#pragma unroll
  for (int i = 0; i < 4; ++i) {
    float4 v = w[i];
    atomicAdd(&hd[4 * i + 0], v.x * iv);
    atomicAdd(&hd[4 * i + 1], v.y * iv);
    atomicAdd(&hd[4 * i + 2], v.z * iv);
    atomicAdd(&hd[4 * i + 3], v.w * iv);
  }
}

// ---------------- kernel: ReLU in place ----------------
__global__ void k_relu(float* __restrict__ h, int n) {
  int i = blockIdx.x * blockDim.x + threadIdx.x;
  if (i >= n) return;
  float v = h[i];
  h[i] = v > 0.0f ? v : 0.0f;
}

// ---- kernel: out = bias2 + h @ root2 via V_WMMA_F32_16X16X4_F32 -------------
// One wave per 16-row tile of h; K=16 as 4 chained WMMAs.
// B (root2 zero-padded to 16x16) is staged once per block in LDS as
// interleaved (K, K+1) float2 pairs: each lane's per-step B operand is a
// single unconditional ds_load_b64 -> no EXEC masking inside the WMMA loop.
__global__ void k_gemm_root2(const float* __restrict__ h, const float* __restrict__ root2,
                             const float* __restrict__ bias2, float* __restrict__ out,
                             int nTiles) {
  __shared__ float2 bpair[8 * 16];  // [kb/2][n] = { B[kb][n], B[kb+1][n] }, kb even
  for (int i = threadIdx.x; i < 8 * 16; i += blockDim.x) {
    int kp = i >> 4;          // pair index: rows 2*kp, 2*kp+1
    int col = i & 15;         // N column
    float2 v;
    v.x = (col < CLS) ? root2[(2 * kp) * CLS + col] : 0.0f;
    v.y = (col < CLS) ? root2[(2 * kp + 1) * CLS + col] : 0.0f;
    bpair[i] = v;
  }
  __syncthreads();

  int lane = threadIdx.x & 31;
  int wave = (blockIdx.x * blockDim.x + threadIdx.x) >> 5;
  if (wave >= nTiles) return;           // wave-uniform: EXEC all-1s for WMMAs
  int half = lane >> 4;                 // 0: lanes 0-15, 1: lanes 16-31
  int l = lane & 15;
  int rowBase = wave * 16;
  int m = rowBase + l;                  // A-matrix row for this lane

  v8f c = {0.f, 0.f, 0.f, 0.f, 0.f, 0.f, 0.f, 0.f};
#pragma unroll
  for (int k0 = 0; k0 < HID; k0 += 4) {
    int kb = k0 + 2 * half;             // even by construction
    // A (16x4 f32): VGPR0 = K=kb, VGPR1 = K=kb+1 for this lane
    v2f a = *reinterpret_cast<const v2f*>(&h[(size_t)m * HID + kb]);
    // B (4x16 f32): one 8B LDS load gives rows (kb, kb+1) at column l
    float2 bv = bpair[(kb >> 1) * 16 + l];
    v2f b; b.x = bv.x; b.y = bv.y;
    c = __builtin_amdgcn_wmma_f32_16x16x4_f32(false, a, false, b, (short)0, c,
                                              false, false);
  }
  // D layout: VGPR i -> row M = i + 8*half, col N = l
  if (l < CLS) {
    float bz = bias2[l];
#pragma unroll
    for (int i = 0; i < 8; ++i) {
      int row = rowBase + i + 8 * half;
      out[(size_t)row * CLS + l] = c[i] + bz;
    }
  }
}

// scalar tail for N not divisible by 16 (not hit for N=50000, kept for generality)
__global__ void k_gemm_tail(const float* __restrict__ h, const float* __restrict__ root2,
                            const float* __restrict__ bias2, float* __restrict__ out,
                            int n0, int N) {
  int n = n0 + blockIdx.x * blockDim.x + threadIdx.x;
  if (n >= N) return;
  for (int cc = 0; cc < CLS; ++cc) {
    float acc = bias2[cc];
    for (int f = 0; f < HID; ++f) acc = fmaf(h[(size_t)n * HID + f], root2[f * CLS + cc], acc);
    out[(size_t)n * CLS + cc] = acc;
  }
}

// ---------------- kernel: layer-2 edge pass ----------------
// out[d] += (h[src] @ weight2[r]) * inv[r, d]; weight2 (16KB) cached in LDS
__global__ void k_edge2(const int* __restrict__ src, const int* __restrict__ dst,
                        const int* __restrict__ et, const float* __restrict__ h,
                        const float* __restrict__ w2, const float* __restrict__ inv,
                        float* __restrict__ out, int E, int N, int R) {
  __shared__ float w2s[32 * HID * CLS];   // R<=32 for this problem
  int nW = R * HID * CLS;
  for (int i = threadIdx.x; i < nW; i += blockDim.x) w2s[i] = w2[i];
  __syncthreads();

  int e = blockIdx.x * blockDim.x + threadIdx.x;
  if (e >= E) return;
  int s = src[e], d = dst[e], r = et[e];
  float iv = inv[(size_t)r * N + d];

  const float4* hp = reinterpret_cast<const float4*>(&h[(size_t)s * HID]);
  float4 h0 = hp[0], h1 = hp[1], h2 = hp[2], h3 = hp[3];
  float hs[HID] = {h0.x, h0.y, h0.z, h0.w, h1.x, h1.y, h1.z, h1.w,
                   h2.x, h2.y, h2.z, h2.w, h3.x, h3.y, h3.z, h3.w};
  const float* W = &w2s[r * HID * CLS];

  float y[CLS] = {0.f, 0.f, 0.f, 0.f, 0.f, 0.f, 0.f, 0.f};
#pragma unroll
  for (int f = 0; f < HID; ++f) {
    float hv = hs[f];
#pragma unroll
    for (int cc = 0; cc < CLS; ++cc) y[cc] = fmaf(hv, W[f * CLS + cc], y[cc]);
  }
  float* od = &out[(size_t)d * CLS];
#pragma unroll
  for (int cc = 0; cc < CLS; ++cc) atomicAdd(&od[cc], y[cc] * iv);
}

// ---------------- kernel: in-place log_softmax over 8 classes ----------------
__global__ void k_logsoftmax(float* __restrict__ out, int N) {
  int n = blockIdx.x * blockDim.x + threadIdx.x;
  if (n >= N) return;
  float* p = &out[(size_t)n * CLS];
  float v[CLS];
  float m = -1e30f;
#pragma unroll
  for (int cc = 0; cc < CLS; ++cc) { v[cc] = p[cc]; m = fmaxf(m, v[cc]); }
  float s = 0.0f;
#pragma unroll
  for (int cc = 0; cc < CLS; ++cc) s += __expf(v[cc] - m);
  float lse = m + __logf(s);
#pragma unroll
  for (int cc = 0; cc < CLS; ++cc) p[cc] = v[cc] - lse;
}

extern "C" void kernel_launch(void* const* d_in, const int* in_sizes, int n_in,
                              void* d_out, int out_size, void* d_ws, size_t ws_size,
                              hipStream_t stream) {
  const int*   edge_index = (const int*)d_in[0];
  const int*   edge_type  = (const int*)d_in[1];
  const float* weight1    = (const float*)d_in[2];
  const float* root1      = (const float*)d_in[3];
  const float* bias1      = (const float*)d_in[4];
  const float* weight2    = (const float*)d_in[5];
  const float* root2      = (const float*)d_in[6];
  const float* bias2      = (const float*)d_in[7];
  float* out = (float*)d_out;

  const int E  = in_sizes[1];
  const int Hn = in_sizes[4];                 // 16
  const int C  = in_sizes[7];                 // 8
  const int N  = in_sizes[3] / Hn;
  const int R  = in_sizes[5] / (Hn * C);

  // workspace: inv (R*N f32, first used as u32 counts), then h (N*16 f32)
  float* inv = (float*)d_ws;
  float* h   = inv + (size_t)R * N;

  const int* src = edge_index;
  const int* dst = edge_index + E;

  const int RN = R * N;
  const int NH = N * HID;
  dim3 blk(256);

  k_zero_u32<<<(RN + 255) / 256, blk, 0, stream>>>((unsigned*)inv, RN);
  k_count<<<(E + 255) / 256, blk, 0, stream>>>(dst, edge_type, (unsigned*)inv, E, N);
  k_invcnt<<<(RN + 255) / 256, blk, 0, stream>>>(inv, RN);
  k_init_h<<<(NH + 255) / 256, blk, 0, stream>>>(h, root1, bias1, NH);
  k_edge1<<<(E + 255) / 256, blk, 0, stream>>>(src, dst, edge_type, weight1, inv, h, E, N);
  k_relu<<<(NH + 255) / 256, blk, 0, stream>>>(h, NH);

  int nTiles = N / 16;                        // 3125 full WMMA tiles for N=50000
  if (nTiles > 0)
    k_gemm_root2<<<(nTiles + 7) / 8, blk, 0, stream>>>(h, root2, bias2, out, nTiles);
  if (N - nTiles * 16 > 0)
    k_gemm_tail<<<1, 64, 0, stream>>>(h, root2, bias2, out, nTiles * 16, N);

  k_edge2<<<(E + 255) / 256, blk, 0, stream>>>(src, dst, edge_type, h, weight2, inv, out, E, N, R);
  k_logsoftmax<<<(N + 255) / 256, blk, 0, stream>>>(out, N);
}